// singleGCNLayer_4020089389120
// MI455X (gfx1250) — compile-verified
//
#include <hip/hip_runtime.h>

typedef __attribute__((ext_vector_type(2))) float v2f;
typedef __attribute__((ext_vector_type(8))) float v8f;

#define EMB 256
#define ROWS_PER_BLOCK 64
#define AGG_STRIDE 260   // 256 + 4 pad: A-fragment ds_load_b64 hits distinct banks

// ---------------- utility kernels ----------------

__global__ __launch_bounds__(256) void k_zero(int* __restrict__ p, int n) {
  int i = blockIdx.x * 256 + threadIdx.x;
  if (i < n) p[i] = 0;
}

__global__ __launch_bounds__(256) void k_hist(const int* __restrict__ rows,
                                              int* __restrict__ counts, int E) {
  int e = blockIdx.x * 256 + threadIdx.x;
  if (e < E) atomicAdd(&counts[rows[e]], 1);
}

// per-block exclusive scan (256 elems/block) + block totals
__global__ __launch_bounds__(256) void k_scan1(const int* __restrict__ counts,
                                               int* __restrict__ excl,
                                               int* __restrict__ blockSums, int n) {
  __shared__ int tmp[256];
  int tid = threadIdx.x;
  int gid = blockIdx.x * 256 + tid;
  int v = (gid < n) ? counts[gid] : 0;
  tmp[tid] = v;
  __syncthreads();
  for (int off = 1; off < 256; off <<= 1) {
    int t = (tid >= off) ? tmp[tid - off] : 0;
    __syncthreads();
    tmp[tid] += t;
    __syncthreads();
  }
  if (gid < n) excl[gid] = tmp[tid] - v;
  if (tid == 255) blockSums[blockIdx.x] = tmp[255];
}

// single-block exclusive scan of block sums (nb <= 512)
__global__ __launch_bounds__(512) void k_scan2(int* __restrict__ blockSums, int nb) {
  __shared__ int tmp[512];
  int tid = threadIdx.x;
  int v = (tid < nb) ? blockSums[tid] : 0;
  tmp[tid] = v;
  __syncthreads();
  for (int off = 1; off < 512; off <<= 1) {
    int t = (tid >= off) ? tmp[tid - off] : 0;
    __syncthreads();
    tmp[tid] += t;
    __syncthreads();
  }
  if (tid < nb) blockSums[tid] = tmp[tid] - v;
}

__global__ __launch_bounds__(256) void k_scan3(int* __restrict__ excl,
                                               const int* __restrict__ blockSums,
                                               int* __restrict__ cursor, int n) {
  int gid = blockIdx.x * 256 + threadIdx.x;
  if (gid < n) {
    int s = excl[gid] + blockSums[blockIdx.x];
    excl[gid] = s;
    cursor[gid] = s;
  }
}

__global__ __launch_bounds__(256) void k_scatter(const int* __restrict__ rows,
                                                 const int* __restrict__ cols,
                                                 const float* __restrict__ vals,
                                                 int* __restrict__ cursor,
                                                 int* __restrict__ ecol,
                                                 float* __restrict__ evals, int E) {
  int e = blockIdx.x * 256 + threadIdx.x;
  if (e < E) {
    int r = rows[e];
    int p = atomicAdd(&cursor[r], 1);
    ecol[p] = cols[e];
    evals[p] = vals[e];
  }
}

// ---------------- fused SpMM (CSR) + WMMA GEMM + ReLU ----------------
// Block: 256 threads = 8 waves, 64 output rows (66.5KB LDS, CDNA5 WGP has 320KB).
// Stage A: wave w reduces rows {8w..8w+7} into LDS aggT (no atomics; whole wave
//          reads each 1KB embedding row as two coalesced 512B instructions).
// Stage B: wave w = (rowTile = w>>1, colHalf = w&1); 2 groups x 4 n-tiles x
//          64 k-steps of V_WMMA_F32_16X16X4_F32; one A ds_load_b64 feeds 4 WMMAs.
__global__ __launch_bounds__(256) void k_fused(const float* __restrict__ emb,
                                               const int* __restrict__ rowStart,
                                               const int* __restrict__ counts,
                                               const int* __restrict__ ecol,
                                               const float* __restrict__ evals,
                                               const float* __restrict__ W,
                                               float* __restrict__ out, int nRows) {
  __shared__ float aggT[ROWS_PER_BLOCK * AGG_STRIDE];   // 66,560 bytes

  const int tid  = threadIdx.x;
  const int wave = tid >> 5;
  const int lane = tid & 31;
  const int lo   = lane & 15;
  const int hi   = lane >> 4;
  const int rowBase = blockIdx.x * ROWS_PER_BLOCK;

  // ---- Stage A: SpMM rows into LDS ----
  for (int j = 0; j < 8; ++j) {
    const int rTile = wave * 8 + j;          // 0..63
    const int r = rowBase + rTile;
    float4 accLo = {0.f, 0.f, 0.f, 0.f};
    float4 accHi = {0.f, 0.f, 0.f, 0.f};
    if (r < nRows) {
      const int s   = rowStart[r];
      const int deg = counts[r];
      for (int jb = 0; jb < deg; jb += 32) {
        // batch-load 32 edges coalesced, broadcast via v_readlane
        int   mc = 0;
        float mv = 0.f;
        if (jb + lane < deg) {
          mc = ecol[s + jb + lane];
          mv = evals[s + jb + lane];
        }
        const int lim = (deg - jb) < 32 ? (deg - jb) : 32;
#pragma unroll 2
        for (int t = 0; t < lim; ++t) {
          const int   c = __builtin_amdgcn_readlane(mc, t);
          const float v =
              __uint_as_float(__builtin_amdgcn_readlane((int)__float_as_uint(mv), t));
          const float4* ep = (const float4*)(emb + (size_t)c * EMB);
          const float4 e0 = ep[lane];        // comps [lane*4, lane*4+4)
          const float4 e1 = ep[lane + 32];   // comps [128+lane*4, ...)
          accLo.x = fmaf(v, e0.x, accLo.x);
          accLo.y = fmaf(v, e0.y, accLo.y);
          accLo.z = fmaf(v, e0.z, accLo.z);
          accLo.w = fmaf(v, e0.w, accLo.w);
          accHi.x = fmaf(v, e1.x, accHi.x);
          accHi.y = fmaf(v, e1.y, accHi.y);
          accHi.z = fmaf(v, e1.z, accHi.z);
          accHi.w = fmaf(v, e1.w, accHi.w);
        }
      }
    }
    float* dst = &aggT[rTile * AGG_STRIDE + lane * 4];
    *(float4*)(dst)       = accLo;
    *(float4*)(dst + 128) = accHi;
  }
  __syncthreads();

  // ---- Stage B: [64 x 256] @ [256 x 256] via f32 WMMA, fused ReLU ----
  const int rowTile = wave >> 1;            // 0..3  -> 16-row tile
  const int colHalf = wave & 1;             // 0..1  -> 128-col half
  const int aRow    = rowTile * 16 + lo;    // A matrix M index within block

  if (rowBase + rowTile * 16 < nRows) {     // full 16-row tiles (N % 16 == 0)
    for (int ng = 0; ng < 2; ++ng) {
      const int nb = colHalf * 128 + ng * 64 + lo;   // base B/D column
      v8f acc[4] = {};
#pragma unroll 8
      for (int ks = 0; ks < 64; ++ks) {
        const int kb = ks * 4 + 2 * hi;
        // A fragment: (A[M=lo][kb], A[M=lo][kb+1]) per ISA 16x4 f32 layout
        v2f a = *(const v2f*)&aggT[aRow * AGG_STRIDE + kb];
#pragma unroll
        for (int nt = 0; nt < 4; ++nt) {
          // B fragment: (B[K=kb][N], B[K=kb+1][N])
          v2f b;
          b.x = W[kb * EMB + nb + nt * 16];
          b.y = W[(kb + 1) * EMB + nb + nt * 16];
          acc[nt] = __builtin_amdgcn_wmma_f32_16x16x4_f32(
              /*neg_a=*/false, a, /*neg_b=*/false, b,
              /*c_mod=*/(short)0, acc[nt], /*reuse_a=*/false, /*reuse_b=*/false);
        }
      }
      // D layout: VGPR r -> row (r + 8*hi), col lo
      const int outRow0 = rowBase + rowTile * 16 + 8 * hi;
#pragma unroll
      for (int nt = 0; nt < 4; ++nt) {
#pragma unroll
        for (int r = 0; r < 8; ++r) {
          float v = acc[nt][r];
          v = v > 0.f ? v : 0.f;
          out[(size_t)(outRow0 + r) * EMB + nb + nt * 16] = v;
        }
      }
    }
  }
}

// ---------------- host launcher ----------------

extern "C" void kernel_launch(void* const* d_in, const int* in_sizes, int n_in,
                              void* d_out, int out_size, void* d_ws, size_t ws_size,
                              hipStream_t stream) {
  const float* emb   = (const float*)d_in[0];
  const int*   arows = (const int*)d_in[1];
  const int*   acols = (const int*)d_in[2];
  const float* avals = (const float*)d_in[3];
  const float* W     = (const float*)d_in[4];
  float* out = (float*)d_out;

  const int N = in_sizes[0] / EMB;   // 100000
  const int E = in_sizes[1];         // 3200000

  // workspace layout (~27 MB)
  int*   counts    = (int*)d_ws;          // N
  int*   rowStart  = counts + N;          // N
  int*   cursor    = rowStart + N;        // N
  int*   blockSums = cursor + N;          // <=512
  int*   ecol      = blockSums + 512;     // E
  float* evals     = (float*)(ecol + E);  // E

  const int nbN = (N + 255) / 256;        // 391 (<=512 for k_scan2)
  const int nbE = (E + 255) / 256;

  k_zero   <<<nbN, 256, 0, stream>>>(counts, N);
  k_hist   <<<nbE, 256, 0, stream>>>(arows, counts, E);
  k_scan1  <<<nbN, 256, 0, stream>>>(counts, rowStart, blockSums, N);
  k_scan2  <<<1,   512, 0, stream>>>(blockSums, nbN);
  k_scan3  <<<nbN, 256, 0, stream>>>(rowStart, blockSums, cursor, N);
  k_scatter<<<nbE, 256, 0, stream>>>(arows, acols, avals, cursor, ecol, evals, E);

  const int nTiles = (N + ROWS_PER_BLOCK - 1) / ROWS_PER_BLOCK;   // 1563
  k_fused  <<<nTiles, 256, 0, stream>>>(emb, rowStart, counts, ecol, evals, W, out, N);
}